// ScaledDotProductAttention_13881334301212
// MI455X (gfx1250) — compile-verified
//
#include <hip/hip_runtime.h>
#include <hip/hip_bf16.h>

typedef __attribute__((ext_vector_type(2))) float v2f;
typedef __attribute__((ext_vector_type(8))) float v8f;

#define SEQ    2048
#define DH     64
#define BQ     16
#define PAD_S  2052              // score row stride (floats): 2052 % 64 == 4 -> conflict-free A-frag reads
#define NTHREADS 256
#define NWAVES 8

// Fused attention forward producing both output [BH,S,D] and attention [BH,S,S].
// One block per (bh, 16-row q tile). fp32 WMMA (16x16x4) for exact math; the
// kernel is HBM-bound (~905 MB moved vs ~52 GFLOP), so the f32 matrix path is
// optimal. Attention probabilities are written exactly once with a non-temporal
// hint so the 805 MB stream doesn't evict the K/V working set from L2.
__global__ __launch_bounds__(NTHREADS)
void attn_fwd_f32_wmma(const float* __restrict__ Q,
                       const float* __restrict__ K,
                       const float* __restrict__ V,
                       float* __restrict__ Out,
                       float* __restrict__ Attn)
{
    extern __shared__ float smem[];
    float* sc   = smem;                    // [BQ][PAD_S]      scores / probabilities
    float* qs   = sc  + BQ * PAD_S;        // [BQ][DH]         staged Q tile
    float* part = qs  + BQ * DH;           // [4][256]         P*V partials (k-half 1)
    float* red  = part + 4 * 256;          // [16][16]         reduction scratch
    float* rmax = red + 256;               // [16]
    float* rsum = red + 256 + 16;          // [16]  (holds 1/sum)

    const int tid  = threadIdx.x;
    const int lane = tid & 31;
    const int wave = tid >> 5;
    const int ln   = lane & 15;            // 0..15
    const int hi   = lane >> 4;            // 0 or 1 (lane half)

    const int bh = blockIdx.y;
    const int q0 = blockIdx.x * BQ;

    const float* Qb = Q + (size_t)bh * SEQ * DH;
    const float* Kb = K + (size_t)bh * SEQ * DH;
    const float* Vb = V + (size_t)bh * SEQ * DH;
    float* Ob = Out  + (size_t)bh * SEQ * DH;
    float* Ab = Attn + (size_t)bh * SEQ * SEQ;

    // ---- stage Q tile (16x64) into LDS ----
    for (int i = tid; i < BQ * DH; i += NTHREADS)
        qs[i] = Qb[q0 * DH + i];
    __syncthreads();

    // ---- phase 1: scores = (Q K^T) * 1/sqrt(64), 128 column tiles of 16 ----
    // A (16x4 f32): lane L holds row m=L%16, k = vgpr + 2*(L/16)  -> contiguous float2
    // B (4x16 f32): lane L holds col n=L%16, k = vgpr + 2*(L/16)  -> contiguous float2 in K's d-dim
    for (int j = wave; j < SEQ / 16; j += NWAVES) {
        const int n0 = j * 16;
        v8f acc = {};
        #pragma unroll
        for (int d0 = 0; d0 < DH; d0 += 4) {
            v2f a = *(const v2f*)&qs[ln * DH + d0 + 2 * hi];
            v2f b = *(const v2f*)&Kb[(size_t)(n0 + ln) * DH + d0 + 2 * hi];
            acc = __builtin_amdgcn_wmma_f32_16x16x4_f32(false, a, false, b,
                                                        (short)0, acc, false, false);
        }
        // C/D layout: row = r + 8*(lane/16), col = lane%16
        #pragma unroll
        for (int r = 0; r < 8; ++r)
            sc[(r + 8 * hi) * PAD_S + n0 + ln] = acc[r] * 0.125f;
    }
    __syncthreads();

    // ---- phase 2: row max (16 threads per row, chunked) ----
    {
        const int row = tid & 15, chunk = tid >> 4;
        float m = -3.0e38f;
        for (int c = chunk; c < SEQ; c += 16)
            m = fmaxf(m, sc[row * PAD_S + c]);
        red[row * 16 + chunk] = m;
    }
    __syncthreads();
    if (tid < 16) {
        float m = red[tid * 16];
        #pragma unroll
        for (int i = 1; i < 16; ++i) m = fmaxf(m, red[tid * 16 + i]);
        rmax[tid] = m;
    }
    __syncthreads();

    // ---- row sum of exp ----
    {
        const int row = tid & 15, chunk = tid >> 4;
        const float m = rmax[row];
        float s = 0.f;
        for (int c = chunk; c < SEQ; c += 16)
            s += __expf(sc[row * PAD_S + c] - m);
        red[row * 16 + chunk] = s;
    }
    __syncthreads();
    if (tid < 16) {
        float s = 0.f;
        #pragma unroll
        for (int i = 0; i < 16; ++i) s += red[tid * 16 + i];
        rsum[tid] = 1.0f / s;
    }
    __syncthreads();

    // ---- phase 3: normalize in LDS + stream attention to global ----
    // Non-temporal: written once, consumed only from LDS -> keep L2 for K/V reuse.
    for (int idx = tid; idx < BQ * SEQ; idx += NTHREADS) {
        const int row = idx >> 11;          // / 2048
        const int col = idx & (SEQ - 1);
        const float p = __expf(sc[row * PAD_S + col] - rmax[row]) * rsum[row];
        sc[row * PAD_S + col] = p;
        __builtin_nontemporal_store(p, &Ab[(size_t)(q0 + row) * SEQ + col]);
    }
    __syncthreads();

    // ---- phase 4: O = P V.  8 waves = 4 d-tiles x 2 k-halves of 1024 ----
    {
        const int t  = wave & 3;            // d-tile (16 cols of D)
        const int n0 = t * 16;
        const int kh = wave >> 2;           // k-half
        v8f acc = {};
        for (int k0 = kh * (SEQ / 2); k0 < (kh + 1) * (SEQ / 2); k0 += 4) {
            v2f a = *(const v2f*)&sc[ln * PAD_S + k0 + 2 * hi];
            v2f b;
            b.x = Vb[(size_t)(k0 + 2 * hi)     * DH + n0 + ln];
            b.y = Vb[(size_t)(k0 + 2 * hi + 1) * DH + n0 + ln];
            acc = __builtin_amdgcn_wmma_f32_16x16x4_f32(false, a, false, b,
                                                        (short)0, acc, false, false);
        }
        if (kh == 1) {
            #pragma unroll
            for (int r = 0; r < 8; ++r)
                part[t * 256 + r * 32 + lane] = acc[r];
        }
        __syncthreads();
        if (kh == 0) {
            #pragma unroll
            for (int r = 0; r < 8; ++r) {
                const float val = acc[r] + part[t * 256 + r * 32 + lane];
                __builtin_nontemporal_store(val, &Ob[(size_t)(q0 + r + 8 * hi) * DH + n0 + ln]);
            }
        }
    }
}

extern "C" void kernel_launch(void* const* d_in, const int* in_sizes, int n_in,
                              void* d_out, int out_size, void* d_ws, size_t ws_size,
                              hipStream_t stream) {
    const float* Q = (const float*)d_in[0];
    const float* K = (const float*)d_in[1];
    const float* V = (const float*)d_in[2];
    float* out = (float*)d_out;

    const int BH = in_sizes[0] / (SEQ * DH);          // 4*12 = 48
    float* attn = out + (size_t)BH * SEQ * DH;        // tuple: (output, attention) flat

    const size_t smem_bytes =
        (size_t)(BQ * PAD_S + BQ * DH + 4 * 256 + 256 + 32) * sizeof(float);

    dim3 grid(SEQ / BQ, BH);
    attn_fwd_f32_wmma<<<grid, NTHREADS, smem_bytes, stream>>>(Q, K, V, out, attn);
}